// UnivariateOneHotEncodingLayer_18210661335091
// MI455X (gfx1250) — compile-verified
//
#include <hip/hip_runtime.h>
#include <hip/hip_bf16.h>
#include <stdint.h>

// out[n,f] = class_bias[f][idx[n,f]] + global_bias[f]
// N=1048576, F=16, C=1024. Pure gather: ~128 MB streamed @ 23.3 TB/s => ~5.5us.
// CDNA5 path: TDM tensor_load_to_lds stages the 64KB table into LDS in ONE
// DMA instruction per block (TENSORcnt), gathers served from LDS banks.

#define NFEAT   16
#define NCLASS  1024
#define THREADS 256
#define BLOCKS  2048

typedef uint32_t u32x4 __attribute__((ext_vector_type(4)));
typedef uint32_t u32x8 __attribute__((ext_vector_type(8)));
typedef int      i32x4 __attribute__((ext_vector_type(4)));   // native vector: OK for nontemporal builtins
typedef float    f32x4 __attribute__((ext_vector_type(4)));

__global__ __launch_bounds__(THREADS) void onehot_gather_tdm_kernel(
    const int*   __restrict__ idx,         // [N*F] flat, int32
    const float* __restrict__ class_bias,  // [F, C] row-major
    const float* __restrict__ global_bias, // [F]
    float*       __restrict__ out,         // [N*F] flat
    int total4)                            // N*F/4
{
    __shared__ float lds_cb[NFEAT * NCLASS];  // 64 KB table in LDS

    // ---- Stage the whole 16x1024 f32 tensor into LDS with one TDM op ----
    if (threadIdx.x == 0) {
        uint64_t gaddr    = (uint64_t)(uintptr_t)class_bias;
        uint32_t lds_base = (uint32_t)(uintptr_t)&lds_cb[0]; // low 32 bits of flat shared ptr == LDS byte offset

        u32x4 g0;
        g0[0] = 1u;                               // count=1, user descriptor, no gather mode
        g0[1] = lds_base;                         // lds_addr (bytes)
        g0[2] = (uint32_t)gaddr;                  // global_addr[31:0]
        g0[3] = ((uint32_t)(gaddr >> 32) & 0x01FFFFFFu) // global_addr[56:32]
              | (2u << 30);                       // type = 2 ("image")

        u32x8 g1;
        g1[0] = 2u << 16;                         // workgroup_mask=0, data_size=2 (4 bytes)
        g1[1] = ((uint32_t)NCLASS & 0xFFFFu) << 16;             // tensor_dim0[15:0] in bits[31:16]
        g1[2] = ((uint32_t)NCLASS >> 16) | ((uint32_t)NFEAT << 16); // dim0[31:16] | tensor_dim1[15:0]
        g1[3] = ((uint32_t)NFEAT  >> 16) | ((uint32_t)NCLASS << 16); // dim1[31:16] | tile_dim0
        g1[4] = (uint32_t)NFEAT;                  // tile_dim1=16, tile_dim2=0 (2D)
        g1[5] = (uint32_t)NCLASS;                 // tensor_dim0_stride[31:0] = 1024 elements
        g1[6] = 0u;                               // stride0[47:32]=0 | dim1_stride[15:0]=0 (unused, 2D)
        g1[7] = 0u;

        // 2-group form: tensors up to 2D (VADDR2/VADDR3 = NULL)
        asm volatile("tensor_load_to_lds %0, %1" : : "s"(g0), "s"(g1) : "memory");
        __builtin_amdgcn_s_wait_tensorcnt(0);     // DMA complete -> LDS data valid
    }
    __syncthreads();                              // publish LDS to all waves

    // ---- Streaming gather: 4 elements (one feature quad) per thread/iter ----
    const i32x4* __restrict__ in4  = (const i32x4*)idx;
    f32x4*       __restrict__ out4 = (f32x4*)out;

    const int stride = (int)(gridDim.x * blockDim.x);   // multiple of 4
    int k = (int)(blockIdx.x * blockDim.x + threadIdx.x);

    const int f0 = (k & 3) * 4;                   // loop-invariant feature base {0,4,8,12}
    const float b0 = global_bias[f0 + 0];
    const float b1 = global_bias[f0 + 1];
    const float b2 = global_bias[f0 + 2];
    const float b3 = global_bias[f0 + 3];
    const float* __restrict__ row = &lds_cb[f0 * NCLASS];

    for (; k < total4; k += stride) {
        __builtin_prefetch(&in4[k + stride], 0, 0);              // global_prefetch_b8
        i32x4 v = __builtin_nontemporal_load(&in4[k]);           // coalesced 16B, TH_NT
        f32x4 r;
        r.x = row[0 * NCLASS + v.x] + b0;                        // ds_load_b32 gathers
        r.y = row[1 * NCLASS + v.y] + b1;
        r.z = row[2 * NCLASS + v.z] + b2;
        r.w = row[3 * NCLASS + v.w] + b3;
        __builtin_nontemporal_store(r, &out4[k]);                // coalesced 16B, TH_NT
    }
}

extern "C" void kernel_launch(void* const* d_in, const int* in_sizes, int n_in,
                              void* d_out, int out_size, void* d_ws, size_t ws_size,
                              hipStream_t stream) {
    const int*   idx = (const int*)  d_in[0];   // inputs  [N, F] (int)
    const float* cb  = (const float*)d_in[1];   // class_bias [F, C]
    const float* gb  = (const float*)d_in[2];   // global_bias [F]
    float*       out = (float*)d_out;           // [N, F]

    const int total4 = out_size / 4;            // 16M / 4 = 4M quads
    onehot_gather_tdm_kernel<<<BLOCKS, THREADS, 0, stream>>>(idx, cb, gb, out, total4);
}